// MACELayer_34342558498838
// MI455X (gfx1250) — compile-verified
//
#include <hip/hip_runtime.h>
#include <hip/hip_bf16.h>
#include <math.h>
#include <stdint.h>

// ---------------- problem constants ----------------
#define NN     20000
#define FFC    64
#define ZZ     10
#define EE     320000
#define RDD    8
#define EPSC   0.25f
#define INVF   0.125f                 // 1/sqrt(64)
#define SQRT3F 1.7320508075688772f

#define USE_ASYNC 1                   // gfx1250 async global->LDS staging
#define GEMM_GT   8                   // row tiles per persistent GEMM block

typedef __attribute__((ext_vector_type(2))) float v2f;
typedef __attribute__((ext_vector_type(8))) float v8f;

__device__ __forceinline__ float silu_f(float x) { return x / (1.0f + __expf(-x)); }

// ---------------- WMMA f32 16x16xK tile helpers ----------------
// A fragment (16x4 f32): lanes 0-15 hold row m, K={k0,k0+1}; lanes 16-31 K={k0+2,k0+3}
// B fragment (4x16 f32): lanes hold col n; VGPR0/1 hold rows ka, ka+1
// C/D (16x16 f32): VGPR r -> row r + 8*hi, col = lane&15
template <int K>
__device__ __forceinline__ v8f wmma_acc_f32(const float* __restrict__ Arow,  // A + m*lda
                                            const float* __restrict__ Bcol,  // B + col0 + n
                                            int ldb, v8f c, int hi) {
  #pragma unroll
  for (int k0 = 0; k0 < K; k0 += 4) {
    int ka = k0 + 2 * hi;
    v2f a; a.x = Arow[ka];        a.y = Arow[ka + 1];
    v2f b; b.x = Bcol[ka * ldb];  b.y = Bcol[(ka + 1) * ldb];
    c = __builtin_amdgcn_wmma_f32_16x16x4_f32(false, a, false, b, (short)0, c,
                                              false, false);
  }
  return c;
}

__device__ __forceinline__ void store_tile16(float* __restrict__ Cbase, int ldc,
                                             v8f c, int lane, float scale) {
  int n = lane & 15, hi = lane >> 4;
  #pragma unroll
  for (int r = 0; r < 8; ++r) Cbase[(r + 8 * hi) * ldc + n] = c[r] * scale;
}

// ---------------- async staging helpers ----------------
__device__ __forceinline__ void async_wait0() {
#if USE_ASYNC
  asm volatile("s_wait_asynccnt 0x0" ::: "memory");
#endif
}

// stage one 16x64 A tile (rows row0..row0+15) into sA[16][68]; 2 x b128 per thread
__device__ __forceinline__ void stage_tileA(const float* __restrict__ A, uint64_t Abase,
                                            int row0, float (*sA)[68], int tid) {
  #pragma unroll
  for (int u = 0; u < 2; ++u) {
    int c  = tid + u * 128;            // 0..255 chunks of 4 floats
    int r  = c >> 4;
    int c4 = (c & 15) * 4;
#if USE_ASYNC
    unsigned goff  = (unsigned)(((row0 + r) * 64 + c4) * 4);
    unsigned laddr = (unsigned)(uintptr_t)(&sA[r][c4]);   // LDS offset = addr[31:0]
    asm volatile("global_load_async_to_lds_b128 %0, %1, %2"
                 :: "v"(laddr), "v"(goff), "s"(Abase) : "memory");
#else
    *(float4*)&sA[r][c4] = *(const float4*)(A + (size_t)(row0 + r) * 64 + c4);
#endif
  }
}

// ---- persistent-block C[M,64] = scale * A[M,64] @ B[64,64]:
// B fragments in registers, A tiles double-buffered through LDS (async DMA) ----
__global__ void __launch_bounds__(128)
k_gemm64(const float* __restrict__ A, const float* __restrict__ B,
         float* __restrict__ C, int M, float scale) {
  __shared__ float sA[2][16][68];
  int tid = threadIdx.x, wave = tid >> 5, lane = tid & 31;
  int m = lane & 15, hi = lane >> 4;
  int rowbase = blockIdx.x * (16 * GEMM_GT);
  if (rowbase >= M) return;
  int ntiles = (M - rowbase) >> 4;
  if (ntiles > GEMM_GT) ntiles = GEMM_GT;

  // hoist this wave's B column-slice fragments (64x16) into registers
  float bx[16], by[16];
  const float* Bcol = B + wave * 16 + m;
  #pragma unroll
  for (int k = 0; k < 16; ++k) {
    int ka = 4 * k + 2 * hi;
    bx[k] = Bcol[ka * 64];
    by[k] = Bcol[(ka + 1) * 64];
  }

  uint64_t Abase = (uint64_t)(uintptr_t)A;
  stage_tileA(A, Abase, rowbase, sA[0], tid);
  async_wait0();
  __syncthreads();

  for (int g = 0; g < ntiles; ++g) {
    int buf = g & 1;
    if (g + 1 < ntiles) stage_tileA(A, Abase, rowbase + 16 * (g + 1), sA[buf ^ 1], tid);
    const float* Ar = &sA[buf][m][0];
    v8f c = {};
    #pragma unroll
    for (int k = 0; k < 16; ++k) {
      int ka = 4 * k + 2 * hi;
      v2f a; a.x = Ar[ka]; a.y = Ar[ka + 1];
      v2f b; b.x = bx[k];  b.y = by[k];
      c = __builtin_amdgcn_wmma_f32_16x16x4_f32(false, a, false, b, (short)0, c,
                                                false, false);
    }
    store_tile16(C + (size_t)(rowbase + 16 * g) * 64 + wave * 16, 64, c, lane, scale);
    async_wait0();       // next buffer staged
    __syncthreads();     // all waves done reading current buffer
  }
}

// ------------- species-indexed GEMM: C[r] = scale * A[r] @ Bz[sidx[r/rpn]] -------------
__global__ void k_gemm64_species(const float* __restrict__ A, const float* __restrict__ Bz,
                                 const int* __restrict__ sidx, int rpn,
                                 float* __restrict__ C, int M, float scale) {
  int wave = threadIdx.x >> 5, lane = threadIdx.x & 31;
  int m = lane & 15, hi = lane >> 4;
  int row0 = blockIdx.x * 16;
  if (row0 >= M) return;
  int s_first = sidx[row0 / rpn];
  int s_last  = sidx[(row0 + 15) / rpn];
  if (s_first == s_last) {                       // uniform tile: WMMA fast path
    const float* B = Bz + s_first * 4096;
    v8f c = {};
    c = wmma_acc_f32<64>(A + (row0 + m) * 64, B + wave * 16 + m, 64, c, hi);
    store_tile16(C + row0 * 64 + wave * 16, 64, c, lane, scale);
  } else {                                       // boundary tile: scalar fallback (rare)
    int t = threadIdx.x;
    for (int u = 0; u < 8; ++u) {
      int idx = t + u * 128;                     // 16 rows x 64 cols
      int r = idx >> 6, col = idx & 63;
      int row = row0 + r;
      const float* B = Bz + sidx[row / rpn] * 4096;
      float acc = 0.f;
      for (int k = 0; k < 64; ++k) acc += A[row * 64 + k] * B[k * 64 + col];
      C[row * 64 + col] = acc * scale;
    }
  }
}

// ---------------- small elementwise kernels ----------------
__global__ void k_sidx(const int* __restrict__ counts, int* __restrict__ sidx) {
  int n = blockIdx.x * blockDim.x + threadIdx.x;
  if (n >= NN) return;
  int acc = 0, s = ZZ - 1;
  for (int z = 0; z < ZZ; ++z) { acc += counts[z]; if (n < acc) { s = z; break; } }
  sidx[n] = s;
}

// node_v (N,F,3) -> v_t (3N, F) contiguous rows
__global__ void k_vt(const float* __restrict__ v, float* __restrict__ vt) {
  int idx = blockIdx.x * blockDim.x + threadIdx.x;
  if (idx >= NN * 192) return;
  int n = idx / 192, rem = idx % 192, f = rem / 3, i = rem % 3;
  vt[(n * 3 + i) * 64 + f] = v[idx];
}

__global__ void k_zero(float* __restrict__ p, int n) {
  int i = blockIdx.x * blockDim.x + threadIdx.x;
  if (i < n) p[i] = 0.f;
}

// wsc_z[z,b,u] = sum_p Wsc[z,p,u] * Proj[p,b]
__global__ void k_wsc(const float* __restrict__ Wsc, const float* __restrict__ Proj,
                      float* __restrict__ out) {
  int idx = blockIdx.x * blockDim.x + threadIdx.x;
  if (idx >= ZZ * 9 * 64) return;
  int z = idx / 576, rem = idx % 576, b = rem / 64, u = rem & 63;
  float acc = 0.f;
  for (int p = 0; p < 9; ++p) acc += Wsc[z * 576 + p * 64 + u] * Proj[p * 9 + b];
  out[z * 576 + b * 64 + u] = acc;
}

// symmetric contraction (correlation 1..3), per (node, channel)
__global__ void k_sym(const float* __restrict__ hs, const float* __restrict__ hvt,
                      const float* __restrict__ wsc, const int* __restrict__ sidx,
                      float* __restrict__ cs, float* __restrict__ cvt) {
  int idx = blockIdx.x * blockDim.x + threadIdx.x;
  if (idx >= NN * 64) return;
  int n = idx >> 6, f = idx & 63;
  const float* w = wsc + sidx[n] * 576 + f;
  float x  = hs[idx];
  float v0 = hvt[(n * 3 + 0) * 64 + f];
  float v1 = hvt[(n * 3 + 1) * 64 + f];
  float v2 = hvt[(n * 3 + 2) * 64 + f];
  float vv = v0 * v0 + v1 * v1 + v2 * v2;
  float x2 = x * x;
  cs[idx] = w[0] * x + w[64] * x2 + w[128] * vv + w[192] * x2 * x + w[256] * x * vv;
  float g = w[320] + w[384] * x + w[448] * x2 + w[512] * vv;
  cvt[(n * 3 + 0) * 64 + f] = g * v0;
  cvt[(n * 3 + 1) * 64 + f] = g * v1;
  cvt[(n * 3 + 2) * 64 + f] = g * v2;
}

// ---------- fused edge kernel: radial MLP (WMMA in LDS) + TP messages + scatter ----------
__global__ void __launch_bounds__(128)
k_edge(const float* __restrict__ re, const float* __restrict__ R1,
       const float* __restrict__ R2, const float* __restrict__ R3,
       const float* __restrict__ R4, const float* __restrict__ vecs,
       const int* __restrict__ senders, const int* __restrict__ receivers,
       const float* __restrict__ hs, const float* __restrict__ hv,
       float* __restrict__ agg_s, float* __restrict__ agg_v) {
  __shared__ float sre[16][8];
  __shared__ float h1[16][68];
  __shared__ float h2[16][68];
  __shared__ float sw[16][256];

  int tid = threadIdx.x;
  int wave = tid >> 5, lane = tid & 31;
  int m = lane & 15, hi = lane >> 4;
  int e0 = blockIdx.x * 16;

  sre[tid >> 3][tid & 7] = re[(e0 + (tid >> 3)) * 8 + (tid & 7)];
  __syncthreads();

  int col0 = wave * 16;
  { // layer1: 8 -> 64, silu
    v8f c = {};
    c = wmma_acc_f32<8>(&sre[m][0], R1 + col0 + m, 64, c, hi);
    #pragma unroll
    for (int r = 0; r < 8; ++r) h1[r + 8 * hi][col0 + m] = silu_f(c[r]);
  }
  __syncthreads();
  { // layer2: 64 -> 64, silu
    v8f c = {};
    c = wmma_acc_f32<64>(&h1[m][0], R2 + col0 + m, 64, c, hi);
    #pragma unroll
    for (int r = 0; r < 8; ++r) h2[r + 8 * hi][col0 + m] = silu_f(c[r]);
  }
  __syncthreads();
  { // layer3: 64 -> 64, silu
    v8f c = {};
    c = wmma_acc_f32<64>(&h2[m][0], R3 + col0 + m, 64, c, hi);
    #pragma unroll
    for (int r = 0; r < 8; ++r) h1[r + 8 * hi][col0 + m] = silu_f(c[r]);
  }
  __syncthreads();
  // layer4: 64 -> 256 (4 column tiles per wave), no activation
  #pragma unroll
  for (int ct = 0; ct < 4; ++ct) {
    int wc0 = (wave * 4 + ct) * 16;
    v8f c = {};
    c = wmma_acc_f32<64>(&h1[m][0], R4 + wc0 + m, 256, c, hi);
    #pragma unroll
    for (int r = 0; r < 8; ++r) sw[r + 8 * hi][wc0 + m] = c[r];
  }
  __syncthreads();

  // message phase: 8 threads per edge, 8 channels per thread
  int el = tid >> 3;
  int fg = (tid & 7) * 8;
  int e  = e0 + el;
  int s  = senders[e];
  int rc = receivers[e];
  float vx = vecs[e * 3 + 0], vy = vecs[e * 3 + 1], vz = vecs[e * 3 + 2];
  float rinv = rsqrtf(vx * vx + vy * vy + vz * vz);
  float y0 = SQRT3F * vx * rinv, y1 = SQRT3F * vy * rinv, y2 = SQRT3F * vz * rinv;
  const float is3 = 1.0f / SQRT3F;
  #pragma unroll
  for (int ff = 0; ff < 8; ++ff) {
    int f = fg + ff;
    float es  = hs[s * 64 + f];
    float ev0 = hv[(s * 3 + 0) * 64 + f];
    float ev1 = hv[(s * 3 + 1) * 64 + f];
    float ev2 = hv[(s * 3 + 2) * 64 + f];
    float dot = ev0 * y0 + ev1 * y1 + ev2 * y2;
    float w0 = sw[el][f], w1 = sw[el][64 + f], w2 = sw[el][128 + f], w3 = sw[el][192 + f];
    float ms = w0 * es + w1 * dot * is3;
    float c3 = w3 * es * is3;
    atomicAdd(&agg_s[rc * 64 + f], ms);
    atomicAdd(&agg_v[(rc * 3 + 0) * 64 + f], w2 * ev0 + c3 * y0);
    atomicAdd(&agg_v[(rc * 3 + 1) * 64 + f], w2 * ev1 + c3 * y1);
    atomicAdd(&agg_v[(rc * 3 + 2) * 64 + f], w2 * ev2 + c3 * y2);
  }
}

// ---------------- final: add skips, write h_s / h_v, readout ----------------
__global__ void k_final(const float* __restrict__ cs_lin, const float* __restrict__ skip_s,
                        const float* __restrict__ cv_lin, const float* __restrict__ skip_v,
                        const float* __restrict__ Wr, float* __restrict__ out,
                        float* __restrict__ out_hs, float* __restrict__ out_hv) {
  int wave = threadIdx.x >> 5, lane = threadIdx.x & 31;
  int n = blockIdx.x * 4 + wave;
  if (n >= NN) return;
  float part = 0.f;
  #pragma unroll
  for (int k = 0; k < 2; ++k) {
    int f = lane + 32 * k;
    float h = cs_lin[n * 64 + f] + skip_s[n * 64 + f];
    out_hs[n * 64 + f] = h;
    part += h * Wr[f];
  }
  for (int off = 16; off; off >>= 1) part += __shfl_xor(part, off, 32);
  if (lane == 0) out[n] = part * INVF;
  #pragma unroll
  for (int k = 0; k < 6; ++k) {
    int j = lane + 32 * k;              // over (i*64 + f)
    int i = j >> 6, f = j & 63;
    float h = cv_lin[(n * 3 + i) * 64 + f] + skip_v[(n * 3 + i) * 64 + f];
    out_hv[n * 192 + f * 3 + i] = h;
  }
}

// ---------------- host launcher ----------------
static inline size_t al256(size_t x) { return (x + 255) & ~size_t(255); }

extern "C" void kernel_launch(void* const* d_in, const int* in_sizes, int n_in,
                              void* d_out, int out_size, void* d_ws, size_t ws_size,
                              hipStream_t stream) {
  const float* vectors   = (const float*)d_in[0];
  const float* node_s    = (const float*)d_in[1];
  const float* node_v    = (const float*)d_in[2];
  const float* radial    = (const float*)d_in[3];
  const int*   senders   = (const int*)d_in[4];
  const int*   receivers = (const int*)d_in[5];
  const int*   spcnt     = (const int*)d_in[6];
  const float* Wup0 = (const float*)d_in[7];
  const float* Wup1 = (const float*)d_in[8];
  const float* R1   = (const float*)d_in[9];
  const float* R2   = (const float*)d_in[10];
  const float* R3   = (const float*)d_in[11];
  const float* R4   = (const float*)d_in[12];
  const float* Wd0  = (const float*)d_in[13];
  const float* Wd1  = (const float*)d_in[14];
  const float* Ws0  = (const float*)d_in[15];
  const float* Ws1  = (const float*)d_in[16];
  const float* Wsc  = (const float*)d_in[17];
  const float* Proj = (const float*)d_in[18];
  const float* Wl0  = (const float*)d_in[19];
  const float* Wl1  = (const float*)d_in[20];
  const float* Wr   = (const float*)d_in[21];
  (void)in_sizes; (void)n_in; (void)out_size; (void)ws_size;

  // workspace carve-up (bytes)
  const size_t szS = (size_t)NN * 64 * sizeof(float);        // 5.12 MB
  const size_t szV = (size_t)NN * 3 * 64 * sizeof(float);    // 15.36 MB
  char* p = (char*)d_ws; size_t off = 0;
  int*   sidx    = (int*)(p + off);   off = al256(off + (size_t)NN * sizeof(int));
  float* v_t     = (float*)(p + off); off = al256(off + szV);   // also reused: hv_down
  float* hs_up   = (float*)(p + off); off = al256(off + szS);   // also reused: cs_lin
  float* hv_up   = (float*)(p + off); off = al256(off + szV);   // also reused: cv_lin
  float* skip_s  = (float*)(p + off); off = al256(off + szS);
  float* skip_v  = (float*)(p + off); off = al256(off + szV);
  float* agg_s   = (float*)(p + off); off = al256(off + szS);   // also reused: cs
  float* agg_v   = (float*)(p + off); off = al256(off + szV);   // also reused: cvt (contiguous after agg_s)
  float* hs_down = (float*)(p + off); off = al256(off + szS);
  float* wsc_z   = (float*)(p + off); off = al256(off + (size_t)ZZ * 9 * 64 * sizeof(float));
  float* hv_down = v_t;
  float* cs      = agg_s;
  float* cvt     = agg_v;
  float* cs_lin  = hs_up;
  float* cv_lin  = hv_up;

  float* out    = (float*)d_out;             // (N,1)
  float* out_hs = out + NN;                  // (N,64)
  float* out_hv = out_hs + (size_t)NN * 64;  // (N,64,3)

  const float downsc = EPSC * INVF;
  const int ROWS_PER_BLK = 16 * GEMM_GT;
  const int gN  = (NN + ROWS_PER_BLK - 1) / ROWS_PER_BLK;
  const int g3N = (3 * NN + ROWS_PER_BLK - 1) / ROWS_PER_BLK;

  // 1) species index
  k_sidx<<<(NN + 255) / 256, 256, 0, stream>>>(spcnt, sidx);
  // 2) node_v transpose -> (3N, 64)
  k_vt<<<(NN * 192 + 255) / 256, 256, 0, stream>>>(node_v, v_t);
  // 3) zero aggregation buffers (agg_s and agg_v contiguous: N*64 + 3N*64 = N*256 floats)
  k_zero<<<((NN * 256) + 255) / 256, 256, 0, stream>>>(agg_s, NN * 256);
  // 4) linear_up
  k_gemm64<<<gN, 128, 0, stream>>>(node_s, Wup0, hs_up, NN, INVF);
  k_gemm64<<<g3N, 128, 0, stream>>>(v_t, Wup1, hv_up, 3 * NN, INVF);
  // 5) species skip connections
  k_gemm64_species<<<NN / 16, 128, 0, stream>>>(node_s, Ws0, sidx, 1, skip_s, NN, INVF);
  k_gemm64_species<<<(3 * NN) / 16, 128, 0, stream>>>(v_t, Ws1, sidx, 3, skip_v, 3 * NN, INVF);
  // 6) fused radial MLP + tensor-product messages + scatter
  k_edge<<<EE / 16, 128, 0, stream>>>(radial, R1, R2, R3, R4, vectors, senders,
                                      receivers, hs_up, hv_up, agg_s, agg_v);
  // 7) linear_down (EPS folded into scale)
  k_gemm64<<<gN, 128, 0, stream>>>(agg_s, Wd0, hs_down, NN, downsc);
  k_gemm64<<<g3N, 128, 0, stream>>>(agg_v, Wd1, hv_down, 3 * NN, downsc);
  // 8) projected species weights + symmetric contraction
  k_wsc<<<(ZZ * 9 * 64 + 255) / 256, 256, 0, stream>>>(Wsc, Proj, wsc_z);
  k_sym<<<(NN * 64 + 255) / 256, 256, 0, stream>>>(hs_down, hv_down, wsc_z, sidx, cs, cvt);
  // 9) linear_sc
  k_gemm64<<<gN, 128, 0, stream>>>(cs, Wl0, cs_lin, NN, INVF);
  k_gemm64<<<g3N, 128, 0, stream>>>(cvt, Wl1, cv_lin, 3 * NN, INVF);
  // 10) add skips, readout, write outputs
  k_final<<<(NN + 3) / 4, 128, 0, stream>>>(cs_lin, skip_s, cv_lin, skip_v, Wr,
                                            out, out_hs, out_hv);
}